// AffineTransform_75299366633519
// MI455X (gfx1250) — compile-verified
//
#include <hip/hip_runtime.h>
#include <hip/hip_bf16.h>

#define B_    64
#define F_    256
#define TIN   4096
#define TOUT  4096
#define TPB   256
#define BN_EPS 1e-5f

// ---------------------------------------------------------------------------
// CDNA5 async global->LDS helpers (gfx1250): tracked with ASYNCcnt.
// LDS address operand is the byte offset within the workgroup's LDS
// allocation (hardware adds LDS_BASE), which is exactly the integer value of
// an address-space(3) pointer.
// ---------------------------------------------------------------------------
typedef __attribute__((address_space(3))) void lds_void_t;

__device__ __forceinline__ unsigned lds_off_of(void* p) {
  return (unsigned)(unsigned long long)(lds_void_t*)p;
}

// Default (RT) async copy: allocates in L2 so a later pass can reuse it.
__device__ __forceinline__ void async_load_f4(const float* g, unsigned lds_byte_off) {
  asm volatile("global_load_async_to_lds_b128 %0, %1, off"
               :: "v"(lds_byte_off), "v"((unsigned long long)(const void*)g)
               : "memory");
}

// Non-temporal async copy: last-use read, don't displace useful L2 lines.
__device__ __forceinline__ void async_load_f4_nt(const float* g, unsigned lds_byte_off) {
  asm volatile("global_load_async_to_lds_b128 %0, %1, off th:TH_LOAD_NT"
               :: "v"(lds_byte_off), "v"((unsigned long long)(const void*)g)
               : "memory");
}

__device__ __forceinline__ void wait_async0() {
  asm volatile("s_wait_asynccnt 0" ::: "memory");
}

// Stage one contiguous 4096-float row into LDS with async b128 copies.
// 256 threads x 4 iterations x 16B = 16 KB.
template <bool NT>
__device__ __forceinline__ void stage_row(const float* __restrict__ gbase,
                                          float* rowp, int tid) {
  const unsigned base = lds_off_of(rowp);
#pragma unroll
  for (int k = 0; k < TIN / (TPB * 4); ++k) {
    const int vec = k * TPB + tid;             // which float4
    if (NT) async_load_f4_nt(gbase + (size_t)vec * 4, base + (unsigned)vec * 16u);
    else    async_load_f4   (gbase + (size_t)vec * 4, base + (unsigned)vec * 16u);
  }
  wait_async0();      // this wave's async copies done
  __syncthreads();    // all waves' copies visible
}

// One bilinear resample tap chain, matching the reference math exactly.
__device__ __forceinline__ float resample_one(const float* row, float s,
                                              float off, int t) {
  const float x     = (2.0f * (float)t + 1.0f) * (1.0f / (float)TOUT) - 1.0f;
  const float coord = s * x + off;
  const float ix    = ((coord + 1.0f) * (float)TIN - 1.0f) * 0.5f;
  const float i0f   = floorf(ix);
  const float w     = ix - i0f;
  const int   i0    = (int)i0f;
  const int   i1    = i0 + 1;
  const float v0 = ((unsigned)i0 < (unsigned)TIN) ? row[i0] : 0.0f;
  const float v1 = ((unsigned)i1 < (unsigned)TIN) ? row[i1] : 0.0f;
  return v0 * (1.0f - w) + v1 * w;
}

// ---------------------------------------------------------------------------
// Pass 1: per-(f,b) partial sums of filtered and filtered^2.
// Streams `data` in ~linear address order; rows allocate in L2 (RT).
// ---------------------------------------------------------------------------
__global__ __launch_bounds__(TPB)
void at_stats_kernel(const float* __restrict__ data,
                     const float* __restrict__ translate,
                     const float* __restrict__ scale,
                     float* __restrict__ partials) {
  __shared__ float row[TIN];
  __shared__ float sred[TPB];
  __shared__ float qred[TPB];

  const int f   = blockIdx.x;
  const int b   = blockIdx.y;
  const int tid = threadIdx.x;

  const float s   = expf(scale[f] / 5.0f);
  const float off = s * translate[f];

  const float* gbase = data + ((size_t)b * F_ + f) * TIN;
  stage_row<false>(gbase, row, tid);

  float sm = 0.0f, sq = 0.0f;
#pragma unroll
  for (int k = 0; k < TOUT / TPB; ++k) {
    const int t = tid + k * TPB;               // lane-consecutive -> no LDS bank conflicts
    const float v = resample_one(row, s, off, t);
    sm += v;
    sq += v * v;
  }

  sred[tid] = sm;
  qred[tid] = sq;
  __syncthreads();
  for (int st = TPB / 2; st > 0; st >>= 1) {
    if (tid < st) {
      sred[tid] += sred[tid + st];
      qred[tid] += qred[tid + st];
    }
    __syncthreads();
  }
  if (tid == 0) {
    const size_t idx = 2u * ((size_t)f * B_ + b);
    partials[idx + 0] = sred[0];
    partials[idx + 1] = qred[0];
  }
}

// ---------------------------------------------------------------------------
// Pass 2: reduce 64 partials per feature -> fused BN scale/bias.
// ---------------------------------------------------------------------------
__global__ __launch_bounds__(B_)
void at_finalize_kernel(const float* __restrict__ partials,
                        const float* __restrict__ gamma,
                        const float* __restrict__ beta,
                        float* __restrict__ fparams) {
  __shared__ float sr[B_];
  __shared__ float qr[B_];
  const int f   = blockIdx.x;
  const int tid = threadIdx.x;

  const size_t idx = 2u * ((size_t)f * B_ + tid);
  sr[tid] = partials[idx + 0];
  qr[tid] = partials[idx + 1];
  __syncthreads();
  for (int st = B_ / 2; st > 0; st >>= 1) {
    if (tid < st) {
      sr[tid] += sr[tid + st];
      qr[tid] += qr[tid + st];
    }
    __syncthreads();
  }
  if (tid == 0) {
    const float invN = 1.0f / (float)((size_t)B_ * TOUT);
    const float mean = sr[0] * invN;
    const float var  = qr[0] * invN - mean * mean;
    const float rstd = rsqrtf(var + BN_EPS);
    const float sc   = gamma[f] * rstd;
    const float bs   = beta[f] - mean * sc;
    fparams[f]       = sc;
    fparams[F_ + f]  = bs;
  }
}

// ---------------------------------------------------------------------------
// Pass 3: recompute resample, apply fused BN affine, write output.
// Walks blocks in REVERSE of pass-1 order so its first ~192MB of reads hit
// the data lines pass 1 left resident in L2. Row loads are NT (last use);
// output stores are NT (never re-read) to preserve L2 for the reuse.
// ---------------------------------------------------------------------------
__global__ __launch_bounds__(TPB)
void at_apply_kernel(const float* __restrict__ data,
                     const float* __restrict__ translate,
                     const float* __restrict__ scale,
                     const float* __restrict__ fparams,
                     float* __restrict__ out) {
  __shared__ float row[TIN];

  const int f   = (F_ - 1) - blockIdx.x;       // reverse traversal
  const int b   = (B_ - 1) - blockIdx.y;
  const int tid = threadIdx.x;

  const float s   = expf(scale[f] / 5.0f);
  const float off = s * translate[f];
  const float sc  = fparams[f];
  const float bs  = fparams[F_ + f];

  const float* gbase = data + ((size_t)b * F_ + f) * TIN;
  stage_row<true>(gbase, row, tid);

  float* obase = out + ((size_t)b * F_ + f) * TOUT;
#pragma unroll
  for (int k = 0; k < TOUT / TPB; ++k) {
    const int t = tid + k * TPB;               // coalesced 128B stores per wave
    const float v = resample_one(row, s, off, t);
    __builtin_nontemporal_store(v * sc + bs, &obase[t]);
  }
}

// ---------------------------------------------------------------------------
extern "C" void kernel_launch(void* const* d_in, const int* in_sizes, int n_in,
                              void* d_out, int out_size, void* d_ws, size_t ws_size,
                              hipStream_t stream) {
  (void)in_sizes; (void)n_in; (void)out_size; (void)ws_size;

  const float* data      = (const float*)d_in[0];
  const float* translate = (const float*)d_in[1];
  const float* scale     = (const float*)d_in[2];
  const float* gamma     = (const float*)d_in[3];
  const float* beta      = (const float*)d_in[4];
  float*       out       = (float*)d_out;

  // d_ws layout: [F_*B_*2] partial sums (128 KB) then [2*F_] fused params (2 KB)
  float* partials = (float*)d_ws;
  float* fparams  = partials + (size_t)F_ * B_ * 2;

  dim3 grid(F_, B_);
  at_stats_kernel<<<grid, TPB, 0, stream>>>(data, translate, scale, partials);
  at_finalize_kernel<<<F_, B_, 0, stream>>>(partials, gamma, beta, fparams);
  at_apply_kernel<<<grid, TPB, 0, stream>>>(data, translate, scale, fparams, out);
}